// testnet_big_55439437857238
// MI455X (gfx1250) — compile-verified
//
#include <hip/hip_runtime.h>
#include <hip/hip_bf16.h>
#include <stdint.h>

#define CDIM 256
#define S    65536       // C*C  (p stride between k-rows of w/b)
#define NCH  16777216    // C^3  (batch stride of x)

typedef float        v2f  __attribute__((ext_vector_type(2)));
typedef float        v8f  __attribute__((ext_vector_type(8)));
typedef unsigned int v4u  __attribute__((ext_vector_type(4)));
typedef int          v4i  __attribute__((ext_vector_type(4)));
typedef int          v8i_t __attribute__((ext_vector_type(8)));

// ---------------------------------------------------------------------------
// Kernel 1: bsum[p] = sum_k b[k*S + p] via the f32 WMMA pipe.
// D = A(16x4 = all ones) x B(4x16 = b values) + C  -> every row of D holds the
// column sums. Column index of B/D = lane%16 (documented layout); the K->slot
// mapping is irrelevant for a pure column sum, so this is layout-robust.
// One wave owns 16 p-columns and accumulates 256 k values in 64 WMMAs.
// ---------------------------------------------------------------------------
__global__ __launch_bounds__(256) void bsum_wmma_kernel(const float* __restrict__ b,
                                                        float* __restrict__ bsum) {
    const int tid  = threadIdx.x;
    const int lane = tid & 31;
    const int wave = tid >> 5;
    const int col  = lane & 15;
    const int hi   = lane >> 4;                 // which half of the wave
    const int p    = blockIdx.x * 128 + wave * 16 + col;

    v2f a; a.x = 1.0f; a.y = 1.0f;              // ones matrix (all slots)
    v8f c = {};
    const float* bp = b + p;
    #pragma unroll 4
    for (int t = 0; t < 64; ++t) {
        // lanes 0-15 take k rows {4t,4t+1}; lanes 16-31 take {4t+2,4t+3}
        const int k0 = 4 * t + 2 * hi;
        v2f bv;
        bv.x = bp[(size_t)k0 * S];
        bv.y = bp[(size_t)(k0 + 1) * S];
        c = __builtin_amdgcn_wmma_f32_16x16x4_f32(false, a, false, bv,
                                                  (short)0, c, false, false);
    }
    if (hi == 0) bsum[p] = c[0];                // (M=0, N=col) = column sum
}

// ---------------------------------------------------------------------------
// TDM: DMA an 8-row x 128-col f32 tile of w (row stride S elements) into LDS.
// D# group0/group1 packed per CDNA5 ISA ch.8 (2D tensor, data_size=4B).
// ---------------------------------------------------------------------------
__device__ __forceinline__ void tdm_issue_w_tile(const float* gsrc, uint32_t ldsByteAddr) {
    const uint64_t ga = (uint64_t)(uintptr_t)gsrc;

    v4u g0;
    g0[0] = 1u;                                            // count=1, user desc
    g0[1] = ldsByteAddr;                                   // lds_addr [63:32]
    g0[2] = (uint32_t)(ga & 0xFFFFFFFFu);                  // global_addr lo
    g0[3] = (uint32_t)((ga >> 32) & 0x01FFFFFFu)           // global_addr hi (57b)
          | (2u << 30);                                    // type = 2 ("image")

    v8i_t g1;
    g1[0] = (int)(2u << 16);                               // data_size = 4B
    g1[1] = 0;                                             // tensor_dim0[15:0]=0 (65536)
    g1[2] = (int)((256u << 16) | 0x0001u);                 // tdim0 hi16=1, tdim1 lo16=256
    g1[3] = (int)(128u << 16);                             // tile_dim0 = 128
    g1[4] = 8;                                             // tile_dim1 = 8 rows
    g1[5] = (int)65536u;                                   // tensor_dim0_stride = S
    g1[6] = 0;
    g1[7] = 0;

    v4i g2 = {};                                           // 2D: groups 2/3 zero
    v4i g3 = {};
#if defined(__clang_major__) && (__clang_major__ >= 23)
    v8i_t g4 = {};
    __builtin_amdgcn_tensor_load_to_lds(g0, g1, g2, g3, g4, 0);
#else
    __builtin_amdgcn_tensor_load_to_lds(g0, g1, g2, g3, 0);
#endif
}

// ---------------------------------------------------------------------------
// Kernel 2: gathered dot products.
//   block = 256 threads -> 128 p-sites, 2-way k split (kh = tid>>7).
//   chunk j streams w rows [8j, 8j+8) x 128 cols into LDS via TDM
//   (double buffered, s_wait_tensorcnt); half kh consumes rows 4kh..4kh+3.
//   x is a private sequential float4 stream per lane (4 rows of 1KB).
// ---------------------------------------------------------------------------
__global__ __launch_bounds__(256) void gather_dot_kernel(const float* __restrict__ x,
                                                         const int* __restrict__ tx,
                                                         const int* __restrict__ ty,
                                                         const float* __restrict__ w,
                                                         float* __restrict__ out) {
    __shared__ float wtile[2][8][128];     // 8 KB double-buffered w tile
    __shared__ float sred[256][4];         // 4 KB reduction scratch

    const int tid = threadIdx.x;
    const int pl  = tid & 127;
    const int kh  = tid >> 7;              // 0: k=8j..8j+3, 1: k=8j+4..8j+7
    const int p   = blockIdx.x * 128 + pl;

    const int ra = tx[p];
    const int rc = ty[p];

    const float* xrow = x + (size_t)ra * S + (size_t)rc * CDIM;
    const float4* xr0 = (const float4*)(xrow);
    const float4* xr1 = (const float4*)(xrow + (size_t)1 * NCH);
    const float4* xr2 = (const float4*)(xrow + (size_t)2 * NCH);
    const float4* xr3 = (const float4*)(xrow + (size_t)3 * NCH);

    const float* wBase = w + (size_t)blockIdx.x * 128;

    if (tid == 0) {
        tdm_issue_w_tile(wBase, (uint32_t)(uintptr_t)&wtile[0][0][0]);
    }

    float acc0 = 0.f, acc1 = 0.f, acc2 = 0.f, acc3 = 0.f;

    const int NCHUNK = 32;
    for (int j = 0; j < NCHUNK; ++j) {
        if (tid == 0) {
            if (j + 1 < NCHUNK) {
                tdm_issue_w_tile(wBase + (size_t)(8 * (j + 1)) * S,
                                 (uint32_t)(uintptr_t)&wtile[(j + 1) & 1][0][0]);
                __builtin_amdgcn_s_wait_tensorcnt(1);   // chunk j complete
            } else {
                __builtin_amdgcn_s_wait_tensorcnt(0);
            }
        }
        __syncthreads();                                // tile j visible to all

        const float* wt = &wtile[j & 1][4 * kh][0];
        const float w0 = wt[0   + pl];
        const float w1 = wt[128 + pl];
        const float w2 = wt[256 + pl];
        const float w3 = wt[384 + pl];

        const int fi = 2 * j + kh;                      // float4 index: k=8j+4kh
        const float4 xv0 = xr0[fi];
        const float4 xv1 = xr1[fi];
        const float4 xv2 = xr2[fi];
        const float4 xv3 = xr3[fi];

        acc0 += xv0.x * w0 + xv0.y * w1 + xv0.z * w2 + xv0.w * w3;
        acc1 += xv1.x * w0 + xv1.y * w1 + xv1.z * w2 + xv1.w * w3;
        acc2 += xv2.x * w0 + xv2.y * w1 + xv2.z * w2 + xv2.w * w3;
        acc3 += xv3.x * w0 + xv3.y * w1 + xv3.z * w2 + xv3.w * w3;

        __syncthreads();                                // protect buffer reuse
    }

    sred[tid][0] = acc0; sred[tid][1] = acc1;
    sred[tid][2] = acc2; sred[tid][3] = acc3;
    __syncthreads();

    if (kh == 0) {
        const float bs = out[p];                        // bsum staged by kernel 1
        const float inv = 1.0f / 256.0f;
        #pragma unroll
        for (int n = 0; n < 4; ++n) {
            const float tot = sred[tid][n] + sred[tid + 128][n];
            out[(size_t)n * S + p] = (tot + bs) * inv;  // overwrites bsum for n=0
        }
    }
}

extern "C" void kernel_launch(void* const* d_in, const int* in_sizes, int n_in,
                              void* d_out, int out_size, void* d_ws, size_t ws_size,
                              hipStream_t stream) {
    const float* x  = (const float*)d_in[0];   // (4,256,256,256) f32
    const int*   tx = (const int*)  d_in[1];   // (256,256) i32
    const int*   ty = (const int*)  d_in[2];   // (256,256) i32
    const float* w  = (const float*)d_in[3];   // (256^3,) f32
    const float* b  = (const float*)d_in[4];   // (256^3,) f32
    float*       o  = (float*)d_out;           // (4,1,256,256) f32

    // Stage 1: bsum[p] -> d_out[0:65536]  (WMMA ones-matmul column reduction)
    bsum_wmma_kernel<<<512, 256, 0, stream>>>(b, o);
    // Stage 2: gathered dot products, TDM-streamed w tiles, final scale.
    gather_dot_kernel<<<512, 256, 0, stream>>>(x, tx, ty, w, o);
}